// DNM_Linear_78056735638014
// MI455X (gfx1250) — compile-verified
//
#include <hip/hip_runtime.h>
#include <stdint.h>

// ---- problem constants (match reference) ----
#define K_CONST  0.5f
#define QS_CONST 0.1f
constexpr int Bdim = 512, INdim = 512, OUTdim = 128, Mdim = 8;
constexpr int BT   = 32;            // b-tile per workgroup
constexpr int PAD  = 4;             // float pad per LDS row (bank spread for b128 reads)
constexpr int XROW = INdim + PAD;   // 516 floats per padded row

// ---- CDNA5 async global->LDS (verified lowering in rounds 2/3) ----
#if defined(__has_builtin)
#  if __has_builtin(__builtin_amdgcn_global_load_async_to_lds_b128)
#    define HAVE_ASYNC_LDS 1
#  endif
#  if __has_builtin(__builtin_amdgcn_s_wait_asynccnt)
#    define HAVE_WAIT_ASYNC 1
#  endif
#endif

#if defined(HAVE_ASYNC_LDS)
// Exact parameter types per the round-1 diagnostic: int __vector(4) pointers,
// AS1 (global) source, AS3 (LDS) destination, non-const.
typedef int v4i_t __attribute__((vector_size(16)));
typedef __attribute__((address_space(1))) v4i_t* g_v4i_ptr;
typedef __attribute__((address_space(3))) v4i_t* l_v4i_ptr;
#endif

__global__ __launch_bounds__(256)
void dnm_dendritic_kernel(const float* __restrict__ x,
                          const float* __restrict__ W,
                          const float* __restrict__ q,
                          float* __restrict__ out)
{
    extern __shared__ float smem[];
    float* xs      = smem;                 // BT * XROW floats (padded x tile)
    float* partial = smem + BT * XROW;     // Mdim * BT floats (cross-wave reduce)

    const int o  = blockIdx.x;             // 0..OUT-1
    const int b0 = blockIdx.y * BT;        // b-tile base
    const int t  = threadIdx.x;            // 0..255
    // Wave id == branch index m; readfirstlane makes it provably wave-uniform,
    // which promotes the W/q streams to SMEM (s_load, KMcnt) — verified in round 3.
    const int m  = __builtin_amdgcn_readfirstlane(t >> 5);
    const int lb = t & 31;                 // lane -> local b

    // ---------- stage x tile [BT x IN] into padded LDS (async DMA path) ----------
    // BT*IN/4 = 4096 b128 transfers; 16 per thread.
    const float* xsrc = x + (size_t)b0 * INdim;
#pragma unroll
    for (int k = 0; k < (BT * INdim / 4) / 256; ++k) {
        const int f   = t + k * 256;       // float4 index within tile
        const int row = f >> 7;            // / (IN/4)
        const int col = f & 127;           // % (IN/4)
        const float* src = xsrc + row * INdim + col * 4;
        float*       dst = xs   + row * XROW  + col * 4;
#if defined(HAVE_ASYNC_LDS)
        __builtin_amdgcn_global_load_async_to_lds_b128(
            (g_v4i_ptr)(uintptr_t)src, (l_v4i_ptr)(uintptr_t)dst, 0, 0);
#else
        *(float4*)dst = *(const float4*)src;
#endif
    }
#if defined(HAVE_ASYNC_LDS)
#  if defined(HAVE_WAIT_ASYNC)
    __builtin_amdgcn_s_wait_asynccnt(0);
#  else
    asm volatile("s_wait_asynccnt 0" ::: "memory");
#  endif
#endif
    __syncthreads();

    // ---------- per-thread (b,m) accumulation over i ----------
    // W/q addresses are wave-uniform -> scalar loads; v_fma can consume both
    // scalar operands (s_w, neg s_q) directly: 3 VALU ops per element, no moves.
    const float4* __restrict__ W4 = (const float4*)(W + ((size_t)o * Mdim + m) * INdim);
    const float4* __restrict__ q4 = (const float4*)(q + ((size_t)o * Mdim + m) * INdim);
    __builtin_prefetch(W4, 0, 3);          // -> global_prefetch_b8
    __builtin_prefetch(q4, 0, 3);

    const float* xrow = xs + lb * XROW;

    float acc0 = 0.0f, acc1 = 0.0f, acc2 = 0.0f, acc3 = 0.0f;  // 4 independent chains
#pragma unroll 4
    for (int c = 0; c < INdim / 4; ++c) {
        const float4 xv = *(const float4*)(xrow + 4 * c);  // ds_load_b128 (padded banks)
        const float4 wv = W4[c];                           // SMEM (wave-uniform)
        const float4 qv = q4[c];
        acc0 += fmaxf(fmaf(xv.x, wv.x, -qv.x), 0.0f);
        acc1 += fmaxf(fmaf(xv.y, wv.y, -qv.y), 0.0f);
        acc2 += fmaxf(fmaf(xv.z, wv.z, -qv.z), 0.0f);
        acc3 += fmaxf(fmaf(xv.w, wv.w, -qv.w), 0.0f);
    }
    const float acc = (acc0 + acc1) + (acc2 + acc3);

    // ---------- cross-wave reduction over m (8 waves) ----------
    partial[m * BT + lb] = acc;
    __syncthreads();

    if (m == 0) {
        float s = 0.0f;
#pragma unroll
        for (int mm = 0; mm < Mdim; ++mm) s += partial[mm * BT + lb];
        // relu(K*(xw-q)) = K*relu(xw-q) for K>0; inner relu of a non-negative sum is identity
        const float soma = K_CONST * s;
        out[(size_t)(b0 + lb) * OUTdim + o] =
            fmaxf(K_CONST * (soma - QS_CONST), 0.0f);
    }
}

extern "C" void kernel_launch(void* const* d_in, const int* in_sizes, int n_in,
                              void* d_out, int out_size, void* d_ws, size_t ws_size,
                              hipStream_t stream)
{
    (void)in_sizes; (void)n_in; (void)out_size; (void)d_ws; (void)ws_size;
    const float* x = (const float*)d_in[0];   // [B, IN]
    const float* W = (const float*)d_in[1];   // [OUT, M, IN]
    const float* q = (const float*)d_in[2];   // [OUT, M, IN]
    float* out     = (float*)d_out;           // [B, OUT]

    const dim3 grid(OUTdim, Bdim / BT);       // 128 x 16
    const dim3 block(256);
    const size_t lds_bytes = (size_t)(BT * XROW + Mdim * BT) * sizeof(float); // ~66 KB
    hipLaunchKernelGGL(dnm_dendritic_kernel, grid, block, lds_bytes, stream,
                       x, W, q, out);
}